// PML_5626407158444
// MI455X (gfx1250) — compile-verified
//
#include <hip/hip_runtime.h>
#include <hip/hip_bf16.h>

typedef __attribute__((ext_vector_type(2))) float v2f;
typedef __attribute__((ext_vector_type(8))) float v8f;

namespace {
constexpr int N_PTS  = 8192;
constexpr int N_PARA = 5000;
constexpr int HID    = 50;
constexpr int PTS    = 32;            // points per workgroup
constexpr int BLK    = 256;           // 8 wave32s
constexpr int NWG    = N_PTS / PTS;   // 256 workgroups
constexpr int KPAD   = 52;            // K (=HID) padded to multiple of 4
constexpr int SSTR   = 60;            // state row stride in floats (>=54, ==4 mod 8 for bank spread)
constexpr int WSTR   = 72;            // weight row stride in floats
}

// ---------------------------------------------------------------------------
// Phase 1: forward-mode AD through the MLP for a 32-point tile.
// State S holds 6 channels x 32 points x 52(+pad) floats: rows = ch*32+p.
// Each hidden layer: Z = S @ W  (WMMA f32 16x16x4), then elementwise tanh
// chain recombining the 6 channels. Epilogue computes the 12 quadratic
// moments of (A,B,C,D,E,F) and writes a per-WG partial sum.
// ---------------------------------------------------------------------------
__global__ __launch_bounds__(BLK) void pinn_phase1(
    const float* __restrict__ x,      // [8192][3]
    const float* __restrict__ Win,    // [3][50]
    const float* __restrict__ bin,    // [50]
    const float* __restrict__ Whid,   // [4][50][50]
    const float* __restrict__ bhid,   // [4][50]
    const float* __restrict__ Wout,   // [50][2]
    const float* __restrict__ bout,   // [2]
    float* __restrict__ partial)      // [NWG][16] (first 12 used)
{
    __shared__ float S[6 * PTS * SSTR];   // 46080 B
    __shared__ float Wl[KPAD * WSTR];     // 14976 B
    __shared__ float bl[64];              //   256 B
    __shared__ float outs[PTS * 8];       //  1024 B
    __shared__ float mred[12 * PTS];      //  1536 B   (total 63872 B)

    const int tid  = threadIdx.x;
    const int lane = tid & 31;
    const int wave = tid >> 5;
    const int p0   = blockIdx.x * PTS;

    // ---- input layer: z = x@Win + bin, seed the 6 derivative channels ----
    for (int e = tid; e < PTS * 64; e += BLK) {
        const int p = e >> 6, j = e & 63;
        if (j < KPAD) {
            float h = 0.f, gx = 0.f, gy = 0.f, gt = 0.f, sxx = 0.f, syy = 0.f;
            if (j < HID) {
                const float w0 = Win[j], w1 = Win[50 + j], w2 = Win[100 + j];
                const float xx = x[(p0 + p) * 3 + 0];
                const float xy = x[(p0 + p) * 3 + 1];
                const float xt = x[(p0 + p) * 3 + 2];
                const float a  = fmaf(xx, w0, fmaf(xy, w1, fmaf(xt, w2, bin[j])));
                h = tanhf(a);
                const float t1 = 1.f - h * h;      // tanh'
                const float t2 = -2.f * h * t1;    // tanh''
                gx  = t1 * w0;  gy = t1 * w1;  gt = t1 * w2;
                sxx = t2 * w0 * w0;  syy = t2 * w1 * w1;
            }
            S[(0 * PTS + p) * SSTR + j] = h;
            S[(1 * PTS + p) * SSTR + j] = gx;
            S[(2 * PTS + p) * SSTR + j] = gy;
            S[(3 * PTS + p) * SSTR + j] = gt;
            S[(4 * PTS + p) * SSTR + j] = sxx;
            S[(5 * PTS + p) * SSTR + j] = syy;
        }
    }
    __syncthreads();

    // ---- 4 hidden layers ----
    for (int layer = 0; layer < 4; ++layer) {
        // stage W (zero-padded to 52x64) and bias into LDS
        for (int e = tid; e < KPAD * 64; e += BLK) {
            const int k = e >> 6, n = e & 63;
            Wl[k * WSTR + n] = (k < HID && n < HID) ? Whid[layer * 2500 + k * 50 + n] : 0.f;
        }
        if (tid < 64) bl[tid] = (tid < HID) ? bhid[layer * 50 + tid] : 0.f;
        __syncthreads();

        // GEMM: (192 x 52) @ (52 x 64) -> 12 Mtiles x 4 Ntiles = 48 tiles,
        // 6 tiles per wave, f32 WMMA 16x16x4 along K.
        v8f acc[6];
        const int m_   = lane & 15;
        const int koff = (lane >> 4) << 1;   // 0 for lanes 0-15, 2 for 16-31
#pragma unroll
        for (int t6 = 0; t6 < 6; ++t6) {
            const int t  = wave + t6 * 8;
            const int mt = t >> 2, nt = t & 3;
            v8f c = {0.f, 0.f, 0.f, 0.f, 0.f, 0.f, 0.f, 0.f};
            const float* Srow = &S[(mt * 16 + m_) * SSTR];
            const float* Wcol = &Wl[nt * 16 + m_];
#pragma unroll
            for (int k = 0; k < KPAD; k += 4) {
                const int kk = k + koff;
                v2f a, b;
                a.x = Srow[kk];
                a.y = Srow[kk + 1];
                b.x = Wcol[kk * WSTR];
                b.y = Wcol[(kk + 1) * WSTR];
                c = __builtin_amdgcn_wmma_f32_16x16x4_f32(
                        /*neg_a=*/false, a, /*neg_b=*/false, b,
                        /*c_mod=*/(short)0, c, /*reuse_a=*/false, /*reuse_b=*/false);
            }
            acc[t6] = c;
        }
        __syncthreads();   // all reads of S complete before overwrite

        // write Z back into S (C/D layout: VGPR j -> row j (+8 for hi lanes))
#pragma unroll
        for (int t6 = 0; t6 < 6; ++t6) {
            const int t     = wave + t6 * 8;
            const int mt    = t >> 2, nt = t & 3;
            const int col   = nt * 16 + m_;
            const int rbase = mt * 16 + ((lane >> 4) << 3);
            if (col < SSTR) {
#pragma unroll
                for (int j = 0; j < 8; ++j)
                    S[(rbase + j) * SSTR + col] = acc[t6][j];
            }
        }
        __syncthreads();

        // elementwise tanh chain: recombine the 6 channels per (point, unit)
        for (int e = tid; e < PTS * 64; e += BLK) {
            const int p = e >> 6, j = e & 63;
            if (j < KPAD) {
                const float z   = S[(0 * PTS + p) * SSTR + j] + bl[j];
                const float zx  = S[(1 * PTS + p) * SSTR + j];
                const float zy  = S[(2 * PTS + p) * SSTR + j];
                const float zt  = S[(3 * PTS + p) * SSTR + j];
                const float zxx = S[(4 * PTS + p) * SSTR + j];
                const float zyy = S[(5 * PTS + p) * SSTR + j];
                const float h  = tanhf(z);
                const float t1 = 1.f - h * h;
                const float t2 = -2.f * h * t1;
                S[(0 * PTS + p) * SSTR + j] = h;
                S[(1 * PTS + p) * SSTR + j] = t1 * zx;
                S[(2 * PTS + p) * SSTR + j] = t1 * zy;
                S[(3 * PTS + p) * SSTR + j] = t1 * zt;
                S[(4 * PTS + p) * SSTR + j] = fmaf(t2 * zx, zx, t1 * zxx);
                S[(5 * PTS + p) * SSTR + j] = fmaf(t2 * zy, zy, t1 * zyy);
            }
        }
        __syncthreads();
    }

    // ---- output layer: 8 scalars per point (u,v,u_t,v_t,u_xx,v_xx,u_yy,v_yy)
    {
        const int p = tid >> 3, s = tid & 7;   // 32 pts x 8 scalars == 256 threads
        const int ch   = (s < 2) ? 0 : (s < 4) ? 3 : (s < 6) ? 4 : 5;
        const int colw = s & 1;
        const float* Sr = &S[(ch * PTS + p) * SSTR];
        float dot = 0.f;
        for (int j = 0; j < HID; ++j) dot = fmaf(Sr[j], Wout[j * 2 + colw], dot);
        if (s < 2) dot += bout[colw];
        outs[p * 8 + s] = dot;
    }
    __syncthreads();

    // ---- per-point moment contributions ----
    if (tid < PTS) {
        const float u   = outs[tid * 8 + 0], v   = outs[tid * 8 + 1];
        const float ut  = outs[tid * 8 + 2], vt  = outs[tid * 8 + 3];
        const float uxx = outs[tid * 8 + 4], vxx = outs[tid * 8 + 5];
        const float uyy = outs[tid * 8 + 6], vyy = outs[tid * 8 + 7];
        const float Q = u * u + v * v;
        const float A = vt - 0.5f * uxx - 0.5f * vyy - Q * u + v;   // base1
        const float B = -0.5f * uyy;
        const float C = -Q * v;
        const float D = ut + 0.5f * vxx - 0.5f * uyy + Q * v + u;   // base2
        const float E = 0.5f * vyy;
        const float F = -Q * u;
        mred[ 0 * PTS + tid] = A * A;  mred[ 1 * PTS + tid] = B * B;
        mred[ 2 * PTS + tid] = C * C;  mred[ 3 * PTS + tid] = A * B;
        mred[ 4 * PTS + tid] = A * C;  mred[ 5 * PTS + tid] = B * C;
        mred[ 6 * PTS + tid] = D * D;  mred[ 7 * PTS + tid] = E * E;
        mred[ 8 * PTS + tid] = F * F;  mred[ 9 * PTS + tid] = D * E;
        mred[10 * PTS + tid] = D * F;  mred[11 * PTS + tid] = E * F;
    }
    __syncthreads();
    if (tid < 12) {
        float ssum = 0.f;
        for (int i = 0; i < PTS; ++i) ssum += mred[tid * PTS + i];
        partial[blockIdx.x * 16 + tid] = ssum;
    }
}

// ---------------------------------------------------------------------------
// Reduce per-WG partials into 6 combined coefficients (fixed order: bitwise
// deterministic across graph replays).
// ---------------------------------------------------------------------------
__global__ __launch_bounds__(64) void pinn_reduce(
    const float* __restrict__ partial, float* __restrict__ coef)
{
    __shared__ float M[12];
    const int t = threadIdx.x;
    if (t < 12) {
        float ssum = 0.f;
        for (int w = 0; w < NWG; ++w) ssum += partial[w * 16 + t];
        M[t] = ssum;
    }
    __syncthreads();
    if (t < 6) coef[t] = (M[t] + M[t + 6]) * (1.0f / (float)N_PTS);
}

// ---------------------------------------------------------------------------
// Phase 2: out[p] = K0 + a^2 K1 + c^2 K2 + 2a K3 + 2c K4 + 2ac K5
// ---------------------------------------------------------------------------
__global__ __launch_bounds__(BLK) void pinn_phase2(
    const float* __restrict__ para, const float* __restrict__ coef,
    float* __restrict__ out, int n)
{
    const int p = blockIdx.x * blockDim.x + threadIdx.x;
    if (p < n) {
        const float a = para[p * 3 + 0];
        const float c = para[p * 3 + 2];
        const float r = coef[0]
                      + a * a * coef[1]
                      + c * c * coef[2]
                      + 2.f * a * coef[3]
                      + 2.f * c * coef[4]
                      + 2.f * a * c * coef[5];
        out[p] = r;
    }
}

extern "C" void kernel_launch(void* const* d_in, const int* in_sizes, int n_in,
                              void* d_out, int out_size, void* d_ws, size_t ws_size,
                              hipStream_t stream) {
    const float* x    = (const float*)d_in[0];
    const float* para = (const float*)d_in[1];
    const float* Win  = (const float*)d_in[2];
    const float* bin  = (const float*)d_in[3];
    const float* Whid = (const float*)d_in[4];
    const float* bhid = (const float*)d_in[5];
    const float* Wout = (const float*)d_in[6];
    const float* bout = (const float*)d_in[7];
    float* out = (float*)d_out;

    float* partial = (float*)d_ws;            // NWG*16 floats = 16 KB
    float* coef    = partial + NWG * 16;      // 6 floats

    pinn_phase1<<<NWG, BLK, 0, stream>>>(x, Win, bin, Whid, bhid, Wout, bout, partial);
    pinn_reduce<<<1, 64, 0, stream>>>(partial, coef);
    pinn_phase2<<<(N_PARA + BLK - 1) / BLK, BLK, 0, stream>>>(para, coef, out, N_PARA);
}